// PrototypeAlignmentLoss_57578331570273
// MI455X (gfx1250) — compile-verified
//
#include <hip/hip_runtime.h>
#include <cstdint>
#include <cstddef>

// ---------------------------------------------------------------------------
// PrototypeAlignmentLoss for MI455X (gfx1250, wave32).
// Strategy: avoid the reference's dense 8192x8192 GEMM (69 GFLOP); compute
// coarse assignment (WMMA f32 16x16x4 GEMM + argmin), bucket features by
// cluster deterministically, then per cluster compute only the needed
// 16-feature x 128-proto WMMA tiles (1.6 GFLOP total, L2-resident data).
// Fine stage sliced 8-way per cluster for occupancy (512 blocks, 2048 waves).
// ---------------------------------------------------------------------------

typedef __attribute__((ext_vector_type(2))) float v2f;
typedef __attribute__((ext_vector_type(8))) float v8f;

#define N_FEAT 8192
#define DIM    512
#define KC     64
#define MF     128
#define SLICES 8

static __device__ __forceinline__ v8f wmma_f32_16x16x4(v2f a, v2f b, v8f c) {
    // D = A(16x4 f32) * B(4x16 f32) + C(16x16 f32), full fp32 precision.
    return __builtin_amdgcn_wmma_f32_16x16x4_f32(
        /*neg_a=*/false, a, /*neg_b=*/false, b,
        /*c_mod=*/(short)0, c, /*reuse_a=*/false, /*reuse_b=*/false);
}

// ---- squared L2 norms of rows of X [nrows x DIM], one wave per row --------
__global__ void pal_row_norms(const float* __restrict__ X,
                              float* __restrict__ out, int nrows) {
    int lane = threadIdx.x & 31;
    int row  = blockIdx.x * (blockDim.x >> 5) + (threadIdx.x >> 5);
    if (row >= nrows) return;
    const float* p = X + (size_t)row * DIM;
    float s = 0.f;
    for (int i = lane; i < DIM; i += 32) { float v = p[i]; s += v * v; }
#pragma unroll
    for (int off = 16; off; off >>= 1) s += __shfl_xor(s, off, 32);
    if (lane == 0) out[row] = s;
}

// ---- coarse assignment: argmin_k ||f - c_k||^2 via WMMA f32 GEMM ----------
// One wave handles a 16-feature strip against all K=64 coarse prototypes.
__global__ void pal_coarse_assign(const float* __restrict__ feats,
                                  const float* __restrict__ coarse,
                                  const float* __restrict__ cnorm,
                                  int* __restrict__ assign) {
    int lane  = threadIdx.x & 31;
    int strip = blockIdx.x * (blockDim.x >> 5) + (threadIdx.x >> 5);
    if (strip >= N_FEAT / 16) return;
    int row0 = strip * 16;
    int l15  = lane & 15;
    int half = lane >> 4;

    const float* arow = feats + (size_t)(row0 + l15) * DIM + 2 * half;

    v8f acc[4];
    v8f zero = {0.f, 0.f, 0.f, 0.f, 0.f, 0.f, 0.f, 0.f};
#pragma unroll
    for (int c = 0; c < 4; c++) acc[c] = zero;

    float cn[4];
#pragma unroll
    for (int c = 0; c < 4; c++) cn[c] = cnorm[c * 16 + l15];

    for (int d0 = 0; d0 < DIM; d0 += 4) {
        v2f a = *(const v2f*)(arow + d0);
#pragma unroll
        for (int c = 0; c < 4; c++) {
            v2f b = *(const v2f*)(coarse + (size_t)(c * 16 + l15) * DIM + d0 + 2 * half);
            acc[c] = wmma_f32_16x16x4(a, b, acc[c]);
        }
    }

    // argmin_k (||c_k||^2 - 2 f.c_k)  ==  argmin_k d2 (row norm is constant)
#pragma unroll
    for (int r = 0; r < 8; r++) {
        float best = cn[0] - 2.0f * acc[0][r];
        int   bidx = l15;
#pragma unroll
        for (int c = 1; c < 4; c++) {
            float v   = cn[c] - 2.0f * acc[c][r];
            int   idx = c * 16 + l15;
            if (v < best || (v == best && idx < bidx)) { best = v; bidx = idx; }
        }
#pragma unroll
        for (int off = 1; off < 16; off <<= 1) {
            float ov = __shfl_xor(best, off, 32);
            int   oi = __shfl_xor(bidx, off, 32);
            if (ov < best || (ov == best && oi < bidx)) { best = ov; bidx = oi; }
        }
        if (l15 == 0) assign[row0 + r + 8 * half] = bidx;
    }
}

// ---- deterministic bucketing ----------------------------------------------
__global__ void pal_zero_counts(int* __restrict__ counts) {
    if (threadIdx.x < KC) counts[threadIdx.x] = 0;
}

__global__ void pal_count(const int* __restrict__ assign, int* __restrict__ counts) {
    int n = blockIdx.x * blockDim.x + threadIdx.x;
    if (n < N_FEAT) atomicAdd(&counts[assign[n]], 1);
}

__global__ void pal_prefix(const int* __restrict__ counts, int* __restrict__ offsets) {
    if (threadIdx.x == 0 && blockIdx.x == 0) {
        int s = 0;
        for (int k = 0; k < KC; k++) { offsets[k] = s; s += counts[k]; }
        offsets[KC] = s;
    }
}

// One wave per cluster: ballot-ranked scatter -> deterministic ordering.
__global__ void pal_scatter(const int* __restrict__ assign,
                            const int* __restrict__ offsets,
                            int* __restrict__ order) {
    int k    = blockIdx.x;
    int lane = threadIdx.x & 31;
    int base = offsets[k];
    int cnt  = 0;
    for (int b = 0; b < N_FEAT; b += 32) {
        int n = b + lane;
        int a = assign[n];
        unsigned mm = (unsigned)__ballot(a == k);
        if (a == k) {
            int pos = __popc(mm & ((1u << lane) - 1u));
            order[base + cnt + pos] = n;
        }
        cnt += __popc(mm);
    }
}

// ---- fine stage: per cluster, 16-feature x 128-proto WMMA tiles -----------
// Grid: (KC clusters) x (SLICES tile-slices). Each block: 4 waves.
__global__ void pal_fine_min(const float* __restrict__ feats,
                             const float* __restrict__ fine,
                             const int* __restrict__ order,
                             const int* __restrict__ offsets,
                             const float* __restrict__ fnorm,
                             const float* __restrict__ pnorm,
                             float* __restrict__ partial) {
    __shared__ float wpart[8];
    int k     = blockIdx.x;
    int slice = blockIdx.y;
    int lane  = threadIdx.x & 31;
    int wid   = threadIdx.x >> 5;
    int nw    = blockDim.x >> 5;
    int l15   = lane & 15;
    int half  = lane >> 4;

    int beg    = offsets[k];
    int cnt    = offsets[k + 1] - beg;
    int ntiles = (cnt + 15) >> 4;
    int tstep  = SLICES * nw;                    // global wave stride over tiles
    const float* fbase = fine + (size_t)k * MF * DIM;

    float pn[8];                       // proto norms, invariant per cluster
#pragma unroll
    for (int c = 0; c < 8; c++) pn[c] = pnorm[k * MF + c * 16 + l15];

    float wsum = 0.f;
    for (int t = slice * nw + wid; t < ntiles; t += tstep) {
        int fi  = t * 16 + l15;
        int row = order[beg + ((fi < cnt) ? fi : 0)];   // pad with a valid row
        const float* arow = feats + (size_t)row * DIM + 2 * half;
        float fn = fnorm[row];

        // Prefetch next tile's gathered A rows (global_prefetch_b8).
        int tn = t + tstep;
        if (tn < ntiles) {
            int fin  = tn * 16 + l15;
            int rown = order[beg + ((fin < cnt) ? fin : 0)];
            __builtin_prefetch(feats + (size_t)rown * DIM + 2 * half, 0, 0);
        }

        v8f acc[8];
        v8f zero = {0.f, 0.f, 0.f, 0.f, 0.f, 0.f, 0.f, 0.f};
#pragma unroll
        for (int c = 0; c < 8; c++) acc[c] = zero;

        for (int d0 = 0; d0 < DIM; d0 += 4) {           // 128 iters x 8 WMMAs
            v2f a = *(const v2f*)(arow + d0);
#pragma unroll
            for (int c = 0; c < 8; c++) {
                v2f b = *(const v2f*)(fbase + (size_t)(c * 16 + l15) * DIM + d0 + 2 * half);
                acc[c] = wmma_f32_16x16x4(a, b, acc[c]);
            }
        }

        // d2 = ||f||^2 + ||p||^2 - 2 f.p ; min over the 128 protos of cluster
#pragma unroll
        for (int r = 0; r < 8; r++) {
            float mv = pn[0] - 2.0f * acc[0][r];
#pragma unroll
            for (int c = 1; c < 8; c++) mv = fminf(mv, pn[c] - 2.0f * acc[c][r]);
#pragma unroll
            for (int off = 1; off < 16; off <<= 1)
                mv = fminf(mv, __shfl_xor(mv, off, 32));
            float fnv = __shfl(fn, r + 8 * half, 32);   // fnorm of feature r+8*half
            if (l15 == 0) {
                int fim = t * 16 + r + 8 * half;
                if (fim < cnt) wsum += sqrtf(fmaxf(fnv + mv, 0.f));
            }
        }
    }

    // deterministic wave -> block reduction
#pragma unroll
    for (int off = 16; off; off >>= 1) wsum += __shfl_xor(wsum, off, 32);
    if (lane == 0) wpart[wid] = wsum;
    __syncthreads();
    if (threadIdx.x == 0) {
        float s = 0.f;
        for (int w = 0; w < nw; w++) s += wpart[w];
        partial[k * SLICES + slice] = s;
    }
}

__global__ void pal_finalize(const float* __restrict__ partial, float* __restrict__ out) {
    if (threadIdx.x == 0 && blockIdx.x == 0) {
        float s = 0.f;
        for (int i = 0; i < KC * SLICES; i++) s += partial[i];
        out[0] = s / (float)N_FEAT;
    }
}

extern "C" void kernel_launch(void* const* d_in, const int* in_sizes, int n_in,
                              void* d_out, int out_size, void* d_ws, size_t ws_size,
                              hipStream_t stream) {
    const float* feats  = (const float*)d_in[0];   // [8192, 512]
    const float* coarse = (const float*)d_in[1];   // [64, 512]
    const float* fine   = (const float*)d_in[2];   // [64, 128, 512]
    float* out = (float*)d_out;

    char* ws = (char*)d_ws;
    int*   assign  = (int*)(ws);                               // 8192 ints
    float* fnorm   = (float*)(ws + 4u * 8192);                 // 8192 f32
    float* pnorm   = (float*)(ws + 4u * 16384);                // 8192 f32
    float* cnorm   = (float*)(ws + 4u * 24576);                // 64 f32
    int*   counts  = (int*)(ws + 4u * 24640);                  // 64 ints
    int*   offsets = (int*)(ws + 4u * 24704);                  // 65 ints
    int*   order   = (int*)(ws + 4u * 24770);                  // 8192 ints
    float* partial = (float*)(ws + 4u * (24770 + 8192));       // 512 f32
    // total workspace: ~134 KB

    const int wpb = 8, thr = 256;  // waves per block for norm kernels
    pal_row_norms<<<dim3((N_FEAT + wpb - 1) / wpb), dim3(thr), 0, stream>>>(feats, fnorm, N_FEAT);
    pal_row_norms<<<dim3((KC * MF + wpb - 1) / wpb), dim3(thr), 0, stream>>>(fine, pnorm, KC * MF);
    pal_row_norms<<<dim3((KC + wpb - 1) / wpb), dim3(thr), 0, stream>>>(coarse, cnorm, KC);

    pal_zero_counts<<<dim3(1), dim3(64), 0, stream>>>(counts);
    pal_coarse_assign<<<dim3(128), dim3(128), 0, stream>>>(feats, coarse, cnorm, assign);
    pal_count<<<dim3(N_FEAT / 256), dim3(256), 0, stream>>>(assign, counts);
    pal_prefix<<<dim3(1), dim3(1), 0, stream>>>(counts, offsets);
    pal_scatter<<<dim3(KC), dim3(32), 0, stream>>>(assign, offsets, order);
    pal_fine_min<<<dim3(KC, SLICES), dim3(128), 0, stream>>>(feats, fine, order, offsets,
                                                             fnorm, pnorm, partial);
    pal_finalize<<<dim3(1), dim3(1), 0, stream>>>(partial, out);
}